// GATGraphClassifier_20461224198767
// MI455X (gfx1250) — compile-verified
//
#include <hip/hip_runtime.h>
#include <math.h>

#define HH 8
#define CC 32
#define DD 256          // HH*CC
#define NEG_SLOPE 0.2f

typedef __attribute__((ext_vector_type(2))) float v2f;
typedef __attribute__((ext_vector_type(8))) float v8f;

// ---------------------------------------------------------------------------
// fp32 GEMM via V_WMMA_F32_16X16X4_F32. One wave -> 16x64 C strip (4 tiles):
// each A fragment feeds 4 back-to-back WMMAs with independent accumulators.
// All loads of a k-step are issued BEFORE the 4 WMMAs so the VMEM clause
// overlaps the matrix pipe (staggered s_wait_loadcnt instead of full stalls).
// A: [M,K] row-major, B: [K,N] row-major, C: [M,N]. M%16==0, K%4==0, N%64==0.
// Lane layout (ISA 7.12.2): lanes 0-15 hold K={0,1} (A) / rows M=r (C),
// lanes 16-31 hold K={2,3} (A) / rows M=r+8 (C).
// ---------------------------------------------------------------------------
__global__ void gemm_wmma_f32(const float* __restrict__ A,
                              const float* __restrict__ B,
                              float* __restrict__ C,
                              int M, int K, int N) {
  int wave = (int)((blockIdx.x * blockDim.x + threadIdx.x) >> 5);
  int lane = (int)(threadIdx.x & 31);
  int ngroups = N >> 6;                 // 64-wide output groups
  int tm = wave / ngroups;
  int tg = wave - tm * ngroups;
  int half = lane >> 4;                 // 0: lanes 0-15, 1: lanes 16-31
  int l16  = lane & 15;

  const float* arow  = A + (size_t)(tm * 16 + l16) * K + 2 * half;  // K pair base
  const float* bbase = B + tg * 64 + l16 + (size_t)(2 * half) * N;

  v8f acc[4] = {};
  for (int k = 0; k < K; k += 4) {
    // ---- issue all loads for this k-step first (one VMEM clause) ----
    v2f a = *(const v2f*)(arow + k);                 // K = {0,1} or {2,3}
    const float* bp = bbase + (size_t)k * N;
    v2f b[4];
#pragma unroll
    for (int j = 0; j < 4; ++j) {
      b[j].x = bp[j * 16];                           // row k + 2*half
      b[j].y = bp[j * 16 + N];                       // row k + 2*half + 1
    }
    // ---- then drain them with 4 back-to-back WMMAs ----
#pragma unroll
    for (int j = 0; j < 4; ++j)
      acc[j] = __builtin_amdgcn_wmma_f32_16x16x4_f32(
          /*neg_a=*/false, a, /*neg_b=*/false, b[j],
          /*c_mod=*/(short)0, acc[j], /*reuse_a=*/false, /*reuse_b=*/false);
  }

  float* crow = C + (size_t)(tm * 16 + 8 * half) * N + tg * 64 + l16;
#pragma unroll
  for (int j = 0; j < 4; ++j)
#pragma unroll
    for (int r = 0; r < 8; ++r)
      crow[(size_t)r * N + j * 16] = acc[j][r];      // row M = r + 8*half
}

// ---------------------------------------------------------------------------
// Per-(node,head): alpha_src/alpha_dst dot products + init max/denom.
// ---------------------------------------------------------------------------
__global__ void alpha_init_kernel(const float* __restrict__ h,
                                  const float* __restrict__ a_src,
                                  const float* __restrict__ a_dst,
                                  float* __restrict__ as_, float* __restrict__ ad_,
                                  float* __restrict__ m_, float* __restrict__ den_,
                                  int N) {
  int i = blockIdx.x * blockDim.x + threadIdx.x;   // n*HH + head
  if (i >= N * HH) return;
  int head = i & (HH - 1);
  int n = i >> 3;
  const float4* hp  = (const float4*)(h + (size_t)n * DD + head * CC);
  const float4* asp = (const float4*)(a_src + head * CC);
  const float4* adp = (const float4*)(a_dst + head * CC);
  float s = 0.f, d = 0.f;
#pragma unroll
  for (int c = 0; c < CC / 4; ++c) {
    float4 v = hp[c], va = asp[c], vd = adp[c];
    s += v.x * va.x + v.y * va.y + v.z * va.z + v.w * va.w;
    d += v.x * vd.x + v.y * vd.y + v.z * vd.z + v.w * vd.w;
  }
  as_[i] = s;
  ad_[i] = d;
  m_[i]  = -__builtin_inff();
  den_[i] = 0.f;
}

// IEEE ordering trick: non-negative floats order as signed ints,
// negative floats order inverted as unsigned ints.
__device__ __forceinline__ void atomicMaxF(float* addr, float v) {
  if (v >= 0.f) atomicMax((int*)addr, __float_as_int(v));
  else          atomicMin((unsigned int*)addr, __float_as_uint(v));
}

__device__ __forceinline__ void edge_ends(const int* __restrict__ ei, int e, int E,
                                          int& s, int& d) {
  if (e < E) { s = ei[e]; d = ei[E + e]; }   // edge_index[0,:], edge_index[1,:]
  else       { s = e - E; d = s; }           // appended self loop
}

__device__ __forceinline__ void load8(const float* __restrict__ p, float* v) {
  float4 a = *(const float4*)p;
  float4 b = *(const float4*)(p + 4);
  v[0] = a.x; v[1] = a.y; v[2] = a.z; v[3] = a.w;
  v[4] = b.x; v[5] = b.y; v[6] = b.z; v[7] = b.w;
}

__global__ void edge_max_kernel(const int* __restrict__ ei, int E, int N,
                                const float* __restrict__ as_,
                                const float* __restrict__ ad_,
                                float* __restrict__ m_) {
  int e = blockIdx.x * blockDim.x + threadIdx.x;
  if (e >= E + N) return;
  int s, d; edge_ends(ei, e, E, s, d);
  float xs[8], xd[8];
  load8(as_ + (size_t)s * HH, xs);
  load8(ad_ + (size_t)d * HH, xd);
#pragma unroll
  for (int h = 0; h < HH; ++h) {
    float x = xs[h] + xd[h];
    x = x > 0.f ? x : NEG_SLOPE * x;
    atomicMaxF(&m_[d * HH + h], x);
  }
}

__global__ void edge_den_kernel(const int* __restrict__ ei, int E, int N,
                                const float* __restrict__ as_,
                                const float* __restrict__ ad_,
                                const float* __restrict__ m_,
                                float* __restrict__ den_) {
  int e = blockIdx.x * blockDim.x + threadIdx.x;
  if (e >= E + N) return;
  int s, d; edge_ends(ei, e, E, s, d);
  float xs[8], xd[8], xm[8];
  load8(as_ + (size_t)s * HH, xs);
  load8(ad_ + (size_t)d * HH, xd);
  load8(m_  + (size_t)d * HH, xm);
#pragma unroll
  for (int h = 0; h < HH; ++h) {
    float x = xs[h] + xd[h];
    x = x > 0.f ? x : NEG_SLOPE * x;
    atomicAdd(&den_[d * HH + h], __expf(x - xm[h]));
  }
}

// 4 edges per 256-thread block; 64 threads per edge, each handling 4 channels
// (float4 gather of h[src], 4 scalar f32 atomics into out[dst] -> lands in L2).
__global__ void edge_aggregate_kernel(const int* __restrict__ ei, int E, int N,
                                      const float* __restrict__ h,
                                      const float* __restrict__ as_,
                                      const float* __restrict__ ad_,
                                      const float* __restrict__ m_,
                                      const float* __restrict__ den_,
                                      float* __restrict__ out) {
  int t = threadIdx.x;                       // 0..255
  int e = blockIdx.x * 4 + (t >> 6);
  if (e >= E + N) return;
  int q = t & 63;                            // channels q*4 .. q*4+3
  int s, d; edge_ends(ei, e, E, s, d);
  int head = q >> 3;                         // (q*4)/CC
  float x = as_[(size_t)s * HH + head] + ad_[(size_t)d * HH + head];
  x = x > 0.f ? x : NEG_SLOPE * x;
  float w = __expf(x - m_[(size_t)d * HH + head]) / den_[(size_t)d * HH + head];
  float4 hv = *(const float4*)(h + (size_t)s * DD + q * 4);
  float* op = out + (size_t)d * DD + q * 4;
  atomicAdd(op + 0, hv.x * w);
  atomicAdd(op + 1, hv.y * w);
  atomicAdd(op + 2, hv.z * w);
  atomicAdd(op + 3, hv.w * w);
}

__global__ void bias_relu_kernel(float4* __restrict__ y, const float* __restrict__ b,
                                 size_t n4) {
  size_t i = blockIdx.x * (size_t)blockDim.x + threadIdx.x;
  if (i >= n4) return;
  int c = (int)((i * 4) & (DD - 1));
  float4 v = y[i];
  v.x += b[c];     v.y += b[c + 1]; v.z += b[c + 2]; v.w += b[c + 3];
  v.x = fmaxf(v.x, 0.f); v.y = fmaxf(v.y, 0.f);
  v.z = fmaxf(v.z, 0.f); v.w = fmaxf(v.w, 0.f);
  y[i] = v;
}

__global__ void fill_zero4_kernel(float4* __restrict__ p, size_t n4) {
  size_t i = blockIdx.x * (size_t)blockDim.x + threadIdx.x;
  if (i < n4) p[i] = make_float4(0.f, 0.f, 0.f, 0.f);
}

__global__ void fill_zero_kernel(float* __restrict__ p, size_t n) {
  size_t i = blockIdx.x * (size_t)blockDim.x + threadIdx.x;
  if (i < n) p[i] = 0.f;
}

// One block (64 threads) per node: float4 per thread into pooled[batch[n]].
__global__ void pool_kernel(const float* __restrict__ y, const int* __restrict__ batch,
                            float* __restrict__ pooled, float* __restrict__ cnt, int N) {
  int n = blockIdx.x;
  int q = threadIdx.x;             // 0..63
  if (n >= N) return;
  int g = batch[n];
  float4 v = *(const float4*)(y + (size_t)n * DD + q * 4);
  float* pp = pooled + (size_t)g * DD + q * 4;
  atomicAdd(pp + 0, v.x);
  atomicAdd(pp + 1, v.y);
  atomicAdd(pp + 2, v.z);
  atomicAdd(pp + 3, v.w);
  if (q == 0) atomicAdd(&cnt[g], 1.f);
}

__global__ void fc_kernel(const float* __restrict__ pooled, const float* __restrict__ cnt,
                          const float* __restrict__ Wfc, const float* __restrict__ bfc,
                          float* __restrict__ out, int G, int NC) {
  int i = blockIdx.x * blockDim.x + threadIdx.x;
  if (i >= G * NC) return;
  int g = i / NC, c = i - g * NC;
  float inv = 1.f / fmaxf(cnt[g], 1.f);
  float acc = bfc[c];
  for (int d = 0; d < DD; ++d)
    acc += pooled[(size_t)g * DD + d] * inv * Wfc[d * NC + c];
  out[i] = acc;
}

// ---------------------------------------------------------------------------
extern "C" void kernel_launch(void* const* d_in, const int* in_sizes, int n_in,
                              void* d_out, int out_size, void* d_ws, size_t ws_size,
                              hipStream_t stream) {
  const float* x     = (const float*)d_in[0];
  const int*   ei    = (const int*)d_in[1];
  const int*   batch = (const int*)d_in[2];
  const float* W1  = (const float*)d_in[4];
  const float* a1s = (const float*)d_in[5];
  const float* a1d = (const float*)d_in[6];
  const float* b1  = (const float*)d_in[7];
  const float* W2  = (const float*)d_in[8];
  const float* a2s = (const float*)d_in[9];
  const float* a2d = (const float*)d_in[10];
  const float* b2  = (const float*)d_in[11];
  const float* Wfc = (const float*)d_in[12];
  const float* bfc = (const float*)d_in[13];

  const int N  = in_sizes[2];            // 50000 (multiple of 16)
  const int F  = in_sizes[0] / N;        // 128
  const int E  = in_sizes[1] / 2;        // 800000
  const int G  = 64;                     // N_GRAPHS
  const int NC = in_sizes[12] / DD;      // 10
  const int Etot = E + N;
  const size_t ND = (size_t)N * DD;

  // Workspace layout (fp32): h | y | as | ad | m | den | pooled | cnt
  float* hbuf   = (float*)d_ws;
  float* ybuf   = hbuf + ND;
  float* as_    = ybuf + ND;
  float* ad_    = as_ + (size_t)N * HH;
  float* m_     = ad_ + (size_t)N * HH;
  float* den_   = m_  + (size_t)N * HH;
  float* pooled = den_ + (size_t)N * HH;
  float* cnt    = pooled + (size_t)G * DD;

  // GEMM: (M/16)*(DD/64) waves, 4 waves (128 thr) per block -> exact division
  const int gemm_blocks = (N / 16) * (DD / 64) / 4;
  const int nh_blocks   = (N * HH + 255) / 256;
  const int e_blocks    = (Etot + 255) / 256;
  const int agg_blocks  = (Etot + 3) / 4;
  const int nd4_blocks  = (int)((ND / 4 + 255) / 256);

  auto run_layer = [&](const float* xin, int K, const float* W,
                       const float* asv, const float* adv, const float* bias) {
    gemm_wmma_f32<<<gemm_blocks, 128, 0, stream>>>(xin, W, hbuf, N, K, DD);
    alpha_init_kernel<<<nh_blocks, 256, 0, stream>>>(hbuf, asv, adv, as_, ad_, m_, den_, N);
    edge_max_kernel<<<e_blocks, 256, 0, stream>>>(ei, E, N, as_, ad_, m_);
    edge_den_kernel<<<e_blocks, 256, 0, stream>>>(ei, E, N, as_, ad_, m_, den_);
    fill_zero4_kernel<<<nd4_blocks, 256, 0, stream>>>((float4*)ybuf, ND / 4);
    edge_aggregate_kernel<<<agg_blocks, 256, 0, stream>>>(ei, E, N, hbuf, as_, ad_, m_, den_, ybuf);
    bias_relu_kernel<<<nd4_blocks, 256, 0, stream>>>((float4*)ybuf, bias, ND / 4);
  };

  run_layer(x,    F,  W1, a1s, a1d, b1);   // layer 1: x -> ybuf
  run_layer(ybuf, DD, W2, a2s, a2d, b2);   // layer 2: ybuf -> (hbuf) -> ybuf

  fill_zero_kernel<<<(G * DD + G + 255) / 256, 256, 0, stream>>>(pooled, (size_t)G * DD + G);
  pool_kernel<<<N, 64, 0, stream>>>(ybuf, batch, pooled, cnt, N);
  fc_kernel<<<(G * NC + 255) / 256, 256, 0, stream>>>(pooled, cnt, Wfc, bfc,
                                                      (float*)d_out, G, NC);
}